// BPANSequence_30691836297848
// MI455X (gfx1250) — compile-verified
//
#include <hip/hip_runtime.h>

typedef __attribute__((ext_vector_type(2))) float v2f;
typedef __attribute__((ext_vector_type(8))) float v8f;

#define B_  512
#define L_  64
#define D_  784
#define NE  128
#define NI  32
#define NC  10
#define BL  (B_*L_)      /* 32768 */
#define NPROJ (NE+NI)    /* 160  */

/* workspace layout (float offsets) */
#define WS_WCOMB 0                         /* relu([Wxe;Wxi]) : 160x784 */
#define WS_WEE   (WS_WCOMB + NPROJ*D_)     /* relu(Wee)  128x128 */
#define WS_WIE   (WS_WEE + NE*NE)          /* -relu(Wie) 128x32  */
#define WS_WEI   (WS_WIE + NE*NI)          /* relu(Wei)  32x128  */
#define WS_WII   (WS_WEI + NI*NE)          /* -relu(Wii) 32x32   */
#define WS_XPROJ (WS_WII + NI*NI)          /* xproj [BL][160]    */

/* output layout (float offsets, reference return order) */
#define O_LOGITS 0
#define O_LAST   (B_*L_*NC)
#define O_RE     (O_LAST + B_*NC)
#define O_RI     (O_RE + (size_t)B_*L_*NE)
#define O_BALE   (O_RI + (size_t)B_*L_*NI)
#define O_BALI   (O_BALE + (size_t)B_*L_*NE)

#define WMMA_F32(A, Bf, C) \
    __builtin_amdgcn_wmma_f32_16x16x4_f32(false, (A), false, (Bf), (short)0, (C), false, false)

/* ---------------- weight prep: Dale rectification (+pre-negate inhibitory) */
__global__ void prep_kernel(const float* __restrict__ Wxe, const float* __restrict__ Wxi,
                            const float* __restrict__ Wee, const float* __restrict__ Wie,
                            const float* __restrict__ Wei, const float* __restrict__ Wii,
                            float* __restrict__ ws) {
    int i = blockIdx.x * blockDim.x + threadIdx.x;
    const int nXe = NE*D_, nXi = NI*D_, nEE = NE*NE, nIE = NE*NI, nEI = NI*NE, nII = NI*NI;
    if (i < nXe)               { ws[WS_WCOMB + i]       =  fmaxf(Wxe[i], 0.f); return; }
    i -= nXe;
    if (i < nXi)               { ws[WS_WCOMB + nXe + i] =  fmaxf(Wxi[i], 0.f); return; }
    i -= nXi;
    if (i < nEE)               { ws[WS_WEE + i]         =  fmaxf(Wee[i], 0.f); return; }
    i -= nEE;
    if (i < nIE)               { ws[WS_WIE + i]         = -fmaxf(Wie[i], 0.f); return; }
    i -= nIE;
    if (i < nEI)               { ws[WS_WEI + i]         =  fmaxf(Wei[i], 0.f); return; }
    i -= nEI;
    if (i < nII)               { ws[WS_WII + i]         = -fmaxf(Wii[i], 0.f); return; }
}

/* ---------------- phase 1: xproj[BL,160] = X @ Wcomb.T ---------------------
 * Each wave computes a 16x80 strip: 1 A-frag load feeds 5 WMMAs into 5
 * independent accumulator chains (A reuse + RAW pipelining).               */
#define CG 5
__global__ __launch_bounds__(256) void xproj_kernel(const float* __restrict__ X,
                                                    float* __restrict__ ws) {
    const float* W  = ws + WS_WCOMB;
    float*       XP = ws + WS_XPROJ;
    const int wave = blockIdx.x * (blockDim.x >> 5) + (threadIdx.x >> 5); /* 0..4095 */
    const int lane = threadIdx.x & 31;
    const int half = lane >> 4, nl = lane & 15;
    const int rowTile = wave >> 1;       /* 0..2047 */
    const int cgrp    = wave & 1;        /* 0..1    */
    const int r0 = rowTile * 16;
    const int nbase = cgrp * (CG * 16);  /* 0 or 80 */

    const float* Arow = X + (size_t)(r0 + nl) * D_ + 2*half;
    const float* Brow = W + (size_t)(nbase + nl) * D_ + 2*half;

    v8f acc[CG] = {};
    #pragma unroll 2
    for (int k0 = 0; k0 < D_; k0 += 4) {
        v2f a = *(const v2f*)(Arow + k0);
        #pragma unroll
        for (int c = 0; c < CG; ++c) {
            v2f b = *(const v2f*)(Brow + (size_t)c * 16 * D_ + k0);
            acc[c] = WMMA_F32(a, b, acc[c]);
        }
    }
    #pragma unroll
    for (int c = 0; c < CG; ++c) {
        #pragma unroll
        for (int r = 0; r < 8; ++r)
            XP[(size_t)(r0 + r + 8*half) * NPROJ + nbase + c*16 + nl] = acc[c][r];
    }
}

/* ---------------- phase 2: recurrent EI scan, one WG per 16 batch rows ----- */
__global__ __launch_bounds__(352) void recur_kernel(const float* __restrict__ ws,
                                                    const float* __restrict__ be,
                                                    const float* __restrict__ bi,
                                                    const float* __restrict__ Wro,
                                                    const float* __restrict__ bro,
                                                    float* __restrict__ out) {
    __shared__ float sre[16 * 130];   /* r_e tile, padded stride (bank-safe)  */
    __shared__ float sri[16 * 34];    /* r_i tile */

    const int lane = threadIdx.x & 31;
    const int wid  = threadIdx.x >> 5;          /* 0..10 */
    const int half = lane >> 4, nl = lane & 15;
    const int b0   = blockIdx.x * 16;
    const float* XP = ws + WS_XPROJ;

    for (int i = threadIdx.x; i < 16*130; i += blockDim.x) sre[i] = 0.f;
    for (int i = threadIdx.x; i < 16*34;  i += blockDim.x) sri[i] = 0.f;

    const bool isE = (wid < 8);
    const bool isI = (wid >= 8 && wid < 10);
    const bool isL = (wid == 10);

    /* preload B fragments into registers (fixed across all 64 steps) */
    v2f wb[32];            /* K=128 operand: Wee / Wei / W_ro column tile */
    v2f wn[8];             /* K=32  operand: -Wie / -Wii column tile      */
    v2f zero2 = {0.f, 0.f};
    float bias = 0.f;
    int xcol = 0;

    if (isE) {
        int n = wid*16 + nl;
        const float* Wb = ws + WS_WEE + (size_t)n*NE + 2*half;
        #pragma unroll
        for (int j = 0; j < 32; ++j) wb[j] = *(const v2f*)(Wb + 4*j);
        const float* Wv = ws + WS_WIE + (size_t)n*NI + 2*half;
        #pragma unroll
        for (int j = 0; j < 8; ++j)  wn[j] = *(const v2f*)(Wv + 4*j);
        bias = be[n];
        xcol = n;
    } else if (isI) {
        int n = (wid-8)*16 + nl;
        const float* Wb = ws + WS_WEI + (size_t)n*NE + 2*half;
        #pragma unroll
        for (int j = 0; j < 32; ++j) wb[j] = *(const v2f*)(Wb + 4*j);
        const float* Wv = ws + WS_WII + (size_t)n*NI + 2*half;
        #pragma unroll
        for (int j = 0; j < 8; ++j)  wn[j] = *(const v2f*)(Wv + 4*j);
        bias = bi[n];
        xcol = NE + n;
    } else { /* readout wave: columns 10..15 padded with zeros */
        bool valid = (nl < NC);
        const float* Wb = Wro + (size_t)nl*NE + 2*half;
        #pragma unroll
        for (int j = 0; j < 32; ++j) wb[j] = valid ? *(const v2f*)(Wb + 4*j) : zero2;
        #pragma unroll
        for (int j = 0; j < 8; ++j)  wn[j] = zero2;
        bias = valid ? bro[nl] : 0.f;
    }

    /* software-pipelined x-drive: preload step 0 */
    float xin[8];
    if (!isL) {
        #pragma unroll
        for (int r = 0; r < 8; ++r)
            xin[r] = XP[(size_t)(b0 + r + 8*half) * L_ * NPROJ + xcol];
    }

    v8f st = {};   /* persistent e / i membrane state, C-layout */
    __syncthreads();

    for (int t = 0; t < L_; ++t) {
        v8f acc0 = {}, acc1 = {};
        if (!isL) {
            #pragma unroll
            for (int r = 0; r < 8; ++r) acc0[r] = xin[r];

            /* prefetch next step's x-drive; overlaps the WMMA chain */
            const int tn = (t + 1 < L_) ? t + 1 : t;
            #pragma unroll
            for (int r = 0; r < 8; ++r)
                xin[r] = XP[((size_t)(b0 + r + 8*half) * L_ + tn) * NPROJ + xcol];

            /* += r_e @ W.T (K=128), two interleaved accumulator chains */
            #pragma unroll
            for (int j = 0; j < 32; j += 2) {
                v2f a0 = *(const v2f*)&sre[nl*130 + 4*j     + 2*half];
                v2f a1 = *(const v2f*)&sre[nl*130 + 4*(j+1) + 2*half];
                acc0 = WMMA_F32(a0, wb[j],   acc0);
                acc1 = WMMA_F32(a1, wb[j+1], acc1);
            }
            /* -= r_i @ W.T (K=32, weights pre-negated) */
            #pragma unroll
            for (int j = 0; j < 8; j += 2) {
                v2f a0 = *(const v2f*)&sri[nl*34 + 4*j     + 2*half];
                v2f a1 = *(const v2f*)&sri[nl*34 + 4*(j+1) + 2*half];
                acc0 = WMMA_F32(a0, wn[j],   acc0);
                acc1 = WMMA_F32(a1, wn[j+1], acc1);
            }
        }
        __syncthreads();   /* all LDS A-reads of r(t-1) complete */

        if (!isL) {
            #pragma unroll
            for (int r = 0; r < 8; ++r) {
                int m = r + 8*half;
                size_t bt = (size_t)(b0 + m) * L_ + t;
                float bal = acc0[r] + acc1[r];
                float ns  = 0.5f * (st[r] + bal + bias);   /* ALPHA = 0.5 */
                st[r] = ns;
                float rr = fmaxf(ns, 0.f);
                if (isE) {
                    out[O_BALE + bt*NE + xcol] = bal;
                    out[O_RE   + bt*NE + xcol] = rr;
                    sre[m*130 + xcol] = rr;
                } else {
                    int c = xcol - NE;
                    out[O_BALI + bt*NI + c] = bal;
                    out[O_RI   + bt*NI + c] = rr;
                    sri[m*34 + c] = rr;
                }
            }
        }
        __syncthreads();   /* r(t) visible in LDS */

        if (isL) {
            v8f l0, l1 = {};
            #pragma unroll
            for (int r = 0; r < 8; ++r) l0[r] = bias;
            #pragma unroll
            for (int j = 0; j < 32; j += 2) {
                v2f a0 = *(const v2f*)&sre[nl*130 + 4*j     + 2*half];
                v2f a1 = *(const v2f*)&sre[nl*130 + 4*(j+1) + 2*half];
                l0 = WMMA_F32(a0, wb[j],   l0);
                l1 = WMMA_F32(a1, wb[j+1], l1);
            }
            if (nl < NC) {
                #pragma unroll
                for (int r = 0; r < 8; ++r) {
                    int b = b0 + r + 8*half;
                    size_t bt = (size_t)b * L_ + t;
                    float v = l0[r] + l1[r];
                    out[O_LOGITS + bt*NC + nl] = v;
                    if (t == L_ - 1) out[O_LAST + (size_t)b*NC + nl] = v;
                }
            }
        }
    }
}

extern "C" void kernel_launch(void* const* d_in, const int* in_sizes, int n_in,
                              void* d_out, int out_size, void* d_ws, size_t ws_size,
                              hipStream_t stream) {
    const float* x   = (const float*)d_in[0];
    const float* Wxe = (const float*)d_in[1];
    const float* Wxi = (const float*)d_in[2];
    const float* Wee = (const float*)d_in[3];
    const float* Wie = (const float*)d_in[4];
    const float* Wei = (const float*)d_in[5];
    const float* Wii = (const float*)d_in[6];
    const float* be  = (const float*)d_in[7];
    const float* bi  = (const float*)d_in[8];
    const float* Wro = (const float*)d_in[9];
    const float* bro = (const float*)d_in[10];
    float* out = (float*)d_out;
    float* ws  = (float*)d_ws;

    const int prepN = NE*D_ + NI*D_ + NE*NE + NE*NI + NI*NE + NI*NI;
    prep_kernel<<<(prepN + 255) / 256, 256, 0, stream>>>(Wxe, Wxi, Wee, Wie, Wei, Wii, ws);

    const int waves = (BL/16) * 2;            /* 4096 waves, 8 waves/block */
    xproj_kernel<<<waves / 8, 256, 0, stream>>>(x, ws);

    recur_kernel<<<B_/16, 352, 0, stream>>>(ws, be, bi, Wro, bro, out);
}